// FullyLoss_7834020348690
// MI455X (gfx1250) — compile-verified
//
#include <hip/hip_runtime.h>
#include <stdint.h>

// Problem constants (reference: B=16384, C=4096, K=2048)
#define TPB        256
#define ROWS_PB    16
#define C_DIM      4096
#define K_DIM      2048
#define KPT        (K_DIM / TPB)       // 8 gathered elements per thread per row
#define LOG_CLAMP  (-100.0f)

// ---- CDNA5 async global -> LDS copy (16B per lane), ASYNCcnt tracked ------
// CDNA5 ISA: GLOBAL_LOAD_ASYNC_TO_LDS_B128  vdst(lds addr), vaddr, off
__device__ __forceinline__ void async_b128(const void* g, void* l) {
  uint32_t lds_off = (uint32_t)(uintptr_t)l;   // generic LDS addr[31:0] == LDS offset
  asm volatile("global_load_async_to_lds_b128 %0, %1, off"
               :: "v"(lds_off), "v"(g) : "memory");
}

__device__ __forceinline__ void wait_async_le8() {
  asm volatile("s_wait_asynccnt 8" ::: "memory");
}
__device__ __forceinline__ void wait_async_le0() {
  asm volatile("s_wait_asynccnt 0" ::: "memory");
}

// Issue async copy of one full row (C_DIM floats) of two tensors into LDS.
// 256 threads * 4 floats * 4 chunks = 4096 floats per tensor; 8 issues/thread.
__device__ __forceinline__ void issue_row(const float* __restrict__ out_row,
                                          const float* __restrict__ lab_row,
                                          float* sx, float* sy, int tid) {
#pragma unroll
  for (int ch = 0; ch < C_DIM / (4 * TPB); ++ch) {
    const int off = ch * (4 * TPB) + tid * 4;        // 16B-aligned
    async_b128(out_row + off, sx + off);
    async_b128(lab_row + off, sy + off);
  }
}

__global__ __launch_bounds__(TPB)
void bce_rows_kernel(const float* __restrict__ output,
                     const float* __restrict__ label,
                     const unsigned char* __restrict__ mask,
                     const int* __restrict__ class_idx,
                     float* __restrict__ partial) {
  __shared__ __align__(16) int   s_idx[K_DIM];
  __shared__ __align__(16) float s_x[2][C_DIM];
  __shared__ __align__(16) float s_y[2][C_DIM];
  __shared__ float s_red[TPB];

  const int tid  = threadIdx.x;
  const int row0 = blockIdx.x * ROWS_PB;

  // Active-row list (uniform across block: all lanes read the same bytes).
  int act[ROWS_PB];
  int na = 0;
#pragma unroll
  for (int r = 0; r < ROWS_PB; ++r) {
    if (mask[row0 + r] != 0) act[na++] = row0 + r;
  }

  float acc = 0.0f;
  if (na > 0) {
    // Stage class_idx (8KB): 2 async b128 per thread, issued first (async
    // loads complete in order, so any later wait covers these too).
#pragma unroll
    for (int ch = 0; ch < K_DIM / (4 * TPB); ++ch) {
      const int off = ch * (4 * TPB) + tid * 4;
      async_b128(class_idx + off, s_idx + off);
    }
    // Prefetch first active row into buffer 0.
    {
      const long long rb = (long long)act[0] * C_DIM;
      issue_row(output + rb, label + rb, s_x[0], s_y[0], tid);
    }

    int cidx[KPT];   // this thread's gather columns (same for every row)

    for (int i = 0; i < na; ++i) {
      const int  b    = i & 1;
      const bool more = (i + 1 < na);
      if (more) {
        // Double-buffer: prefetch next row while we consume this one.
        const long long rb = (long long)act[i + 1] * C_DIM;
        issue_row(output + rb, label + rb, s_x[1 - b], s_y[1 - b], tid);
        wait_async_le8();   // leave only the 8 just-issued copies in flight
      } else {
        wait_async_le0();
      }
      __syncthreads();      // all waves' portions of buffer b are in LDS

      if (i == 0) {
        // Indices are row-invariant: read once into registers, reuse.
#pragma unroll
        for (int kk = 0; kk < KPT; ++kk) cidx[kk] = s_idx[tid + kk * TPB];
      }

      const float* sx = s_x[b];
      const float* sy = s_y[b];
#pragma unroll
      for (int kk = 0; kk < KPT; ++kk) {
        const int   c = cidx[kk];
        const float x = sx[c];
        const float y = sy[c];
        // log1p(-x) == log(1-x) to ulp accuracy on (0,1): 1-x is exact for
        // x>=0.5 (Sterbenz) and |log(1-x)| >= log(2) for x<0.5.
        const float lx = fmaxf(__logf(x),        LOG_CLAMP);
        const float l1 = fmaxf(__logf(1.0f - x), LOG_CLAMP);
        // y*lx + (1-y)*l1 = y*(lx-l1) + l1
        acc += fmaf(y, lx - l1, l1);
      }
      __syncthreads();      // buffer b free for the prefetch two iterations out
    }
  }

  // Deterministic block tree reduction.
  s_red[tid] = acc;
  __syncthreads();
#pragma unroll
  for (int s = TPB / 2; s > 0; s >>= 1) {
    if (tid < s) s_red[tid] += s_red[tid + s];
    __syncthreads();
  }
  if (tid == 0) partial[blockIdx.x] = -s_red[0];   // bce = -(...)
}

__global__ __launch_bounds__(TPB)
void bce_finalize_kernel(const unsigned char* __restrict__ mask, int B,
                         const float* __restrict__ partial, int nblocks,
                         float Kf, float* __restrict__ out) {
  __shared__ float s_f[TPB];
  __shared__ int   s_i[TPB];
  const int tid = threadIdx.x;

  int   ms = 0;
  for (int i = tid; i < B; i += TPB) ms += (mask[i] != 0) ? 1 : 0;
  float ps = 0.0f;
  for (int i = tid; i < nblocks; i += TPB) ps += partial[i];

  s_i[tid] = ms;
  s_f[tid] = ps;
  __syncthreads();
#pragma unroll
  for (int s = TPB / 2; s > 0; s >>= 1) {
    if (tid < s) { s_i[tid] += s_i[tid + s]; s_f[tid] += s_f[tid + s]; }
    __syncthreads();
  }
  if (tid == 0) out[0] = s_f[0] / ((float)s_i[0] * Kf);
}

extern "C" void kernel_launch(void* const* d_in, const int* in_sizes, int n_in,
                              void* d_out, int out_size, void* d_ws, size_t ws_size,
                              hipStream_t stream) {
  const float*         output    = (const float*)d_in[0];
  const float*         label     = (const float*)d_in[1];
  const unsigned char* mask      = (const unsigned char*)d_in[2]; // jax bool = 1 byte
  const int*           class_idx = (const int*)d_in[3];

  const int B  = in_sizes[2];
  const int K  = in_sizes[3];
  const int NB = (B + ROWS_PB - 1) / ROWS_PB;

  float* partial = (float*)d_ws;   // NB floats of scratch

  bce_rows_kernel<<<NB, TPB, 0, stream>>>(output, label, mask, class_idx, partial);
  bce_finalize_kernel<<<1, TPB, 0, stream>>>(mask, B, partial, NB, (float)K,
                                             (float*)d_out);
}